// TransformerSeq2Seq_21930103013720
// MI455X (gfx1250) — compile-verified
//
#include <hip/hip_runtime.h>
#include <hip/hip_bf16.h>
#include <stdint.h>

// Beam-search scorer step (gfx1250 / MI455X).
// Per batch row: stable-partition 16 candidates on (token == eos), keep first 8,
// gather scores/tokens/indices, compute src = local + b*8, mask by done[b].
// input_ids (256 MB) is dead w.r.t. the outputs and is never read.
//
// CDNA5-specific paths used:
//   * global_load_async_to_lds_b128 (ASYNCcnt-tracked async global->LDS DMA)
//   * s_wait_asynccnt split wait counter
//   * wave32 popcount-rank stable partition (no sort)

#define CAND 16           // 2 * num_beams
#define NB   8            // num_beams
#define ROWS_PER_BLOCK 128
#define ROW_STRIDE 80     // 64B payload + 16B pad: keeps 16B align, 2-way LDS banks

__global__ __launch_bounds__(ROWS_PER_BLOCK)
void beam_select_kernel(const float* __restrict__ next_scores,
                        const int*   __restrict__ next_tokens,
                        const int*   __restrict__ next_indices,
                        const unsigned char* __restrict__ done,
                        const int* __restrict__ pad_p,
                        const int* __restrict__ eos_p,
                        float* __restrict__ outS,
                        float* __restrict__ outT,
                        float* __restrict__ outI,
                        int batch)
{
    __shared__ __align__(16) unsigned char smem[3 * ROWS_PER_BLOCK * ROW_STRIDE];

    const int t = threadIdx.x;
    const int b = blockIdx.x * ROWS_PER_BLOCK + t;
    if (b >= batch) return;

    unsigned char* myTok = &smem[(size_t)t * ROW_STRIDE];
    unsigned char* mySc  = &smem[(size_t)(ROWS_PER_BLOCK + t) * ROW_STRIDE];
    unsigned char* myIdx = &smem[(size_t)(2 * ROWS_PER_BLOCK + t) * ROW_STRIDE];

    // addrspacecast(3->0): low 32 bits of the flat shared address == LDS byte offset
    unsigned offTok = (unsigned)(uintptr_t)(void*)myTok;
    unsigned offSc  = (unsigned)(uintptr_t)(void*)mySc;
    unsigned offIdx = (unsigned)(uintptr_t)(void*)myIdx;

    const void* gTok = (const void*)(next_tokens  + (size_t)b * CAND);
    const void* gSc  = (const void*)(next_scores  + (size_t)b * CAND);
    const void* gIdx = (const void*)(next_indices + (size_t)b * CAND);

    // Stage this thread's 3 x 64B row slices into LDS asynchronously.
    // INST_OFFSET is added to BOTH the global and the LDS address, so one
    // base VGPR (pair) per array suffices for the whole 64B copy.
    asm volatile(
        "global_load_async_to_lds_b128 %0, %3, off\n\t"
        "global_load_async_to_lds_b128 %0, %3, off offset:16\n\t"
        "global_load_async_to_lds_b128 %0, %3, off offset:32\n\t"
        "global_load_async_to_lds_b128 %0, %3, off offset:48\n\t"
        "global_load_async_to_lds_b128 %1, %4, off\n\t"
        "global_load_async_to_lds_b128 %1, %4, off offset:16\n\t"
        "global_load_async_to_lds_b128 %1, %4, off offset:32\n\t"
        "global_load_async_to_lds_b128 %1, %4, off offset:48\n\t"
        "global_load_async_to_lds_b128 %2, %5, off\n\t"
        "global_load_async_to_lds_b128 %2, %5, off offset:16\n\t"
        "global_load_async_to_lds_b128 %2, %5, off offset:32\n\t"
        "global_load_async_to_lds_b128 %2, %5, off offset:48"
        :
        : "v"(offTok), "v"(offSc), "v"(offIdx),
          "v"(gTok), "v"(gSc), "v"(gIdx)
        : "memory");

    // Overlap the async DMA with the scalar/flag loads.
    const int pad = *pad_p;
    const int eos = *eos_p;
    const unsigned char dn = done[b];
    const int rowOut = b * NB;

    // Each thread reads only what it requested -> per-wave wait, no barrier.
    asm volatile("s_wait_asynccnt 0" ::: "memory");

    if (dn) {
        const float padf = (float)pad;
        #pragma unroll
        for (int k = 0; k < NB; ++k) {
            outS[rowOut + k] = 0.0f;
            outT[rowOut + k] = padf;
            outI[rowOut + k] = 0.0f;
        }
        return;
    }

    int   tok[CAND];
    float sc [CAND];
    int   idx[CAND];
    #pragma unroll
    for (int q = 0; q < CAND / 4; ++q) {
        int4   t4 = ((const int4*)  myTok)[q];   // ds_load_b128
        float4 s4 = ((const float4*)mySc )[q];
        int4   i4 = ((const int4*)  myIdx)[q];
        tok[4*q+0]=t4.x; tok[4*q+1]=t4.y; tok[4*q+2]=t4.z; tok[4*q+3]=t4.w;
        sc [4*q+0]=s4.x; sc [4*q+1]=s4.y; sc [4*q+2]=s4.z; sc [4*q+3]=s4.w;
        idx[4*q+0]=i4.x; idx[4*q+1]=i4.y; idx[4*q+2]=i4.z; idx[4*q+3]=i4.w;
    }

    // Stable partition via popcount ranks: non-EOS candidates (in order) get
    // ranks [0, totalNon), EOS candidates (in order) follow. Keep rank < 8.
    unsigned eosMask = 0u;
    #pragma unroll
    for (int j = 0; j < CAND; ++j)
        eosMask |= (tok[j] == eos) ? (1u << j) : 0u;
    const unsigned nonMask = (~eosMask) & 0xFFFFu;
    const int totalNon = __popc(nonMask);

    #pragma unroll
    for (int j = 0; j < CAND; ++j) {
        const unsigned below = (1u << j) - 1u;
        const bool isE = (eosMask >> j) & 1u;
        const int rank = isE ? (totalNon + __popc(eosMask & below))
                             : __popc(nonMask & below);
        if (rank < NB) {
            outS[rowOut + rank] = sc[j];
            outT[rowOut + rank] = (float)tok[j];
            outI[rowOut + rank] = (float)(idx[j] + rowOut);  // local + b*NB
        }
    }
}

extern "C" void kernel_launch(void* const* d_in, const int* in_sizes, int n_in,
                              void* d_out, int out_size, void* d_ws, size_t ws_size,
                              hipStream_t stream) {
    (void)n_in; (void)out_size; (void)d_ws; (void)ws_size;

    // setup_inputs order:
    // 0: input_ids (unused), 1: next_scores, 2: next_tokens, 3: next_indices,
    // 4: done (1B bool), 5: pad_token_id, 6: eos_token_id, 7: num_beams (==8)
    const float* next_scores         = (const float*)d_in[1];
    const int*   next_tokens         = (const int*)d_in[2];
    const int*   next_indices        = (const int*)d_in[3];
    const unsigned char* done        = (const unsigned char*)d_in[4];
    const int*   pad_p               = (const int*)d_in[5];
    const int*   eos_p               = (const int*)d_in[6];

    const int batch = in_sizes[1] / CAND;   // next_scores is (batch, 2*NB)

    float* out  = (float*)d_out;            // tuple flattened: scores|tokens|src
    float* outS = out;
    float* outT = out + (size_t)batch * NB;
    float* outI = out + 2 * (size_t)batch * NB;

    dim3 grid((batch + ROWS_PER_BLOCK - 1) / ROWS_PER_BLOCK);
    dim3 block(ROWS_PER_BLOCK);
    beam_select_kernel<<<grid, block, 0, stream>>>(
        next_scores, next_tokens, next_indices, done, pad_p, eos_p,
        outS, outT, outI, batch);
}